// RecurrentAttention_56659208568960
// MI455X (gfx1250) — compile-verified
//
#include <hip/hip_runtime.h>
#include <hip/hip_bf16.h>

// Problem constants (from setup_inputs)
#define B   64
#define T   512
#define D   600
#define DA  300
#define U   300
#define H   5
#define AW_STRIDE (D + DA + U)   // 1200, per-hop stride in al_w
#define NDB 3                    // d-chunks for the streaming kernels
#define DCHUNK (D / NDB)         // 200 floats
#define NQ  (DCHUNK / 4)         // 50 float4 per chunk
#define NT  ((U + 15) / 16)      // 19 N-tiles for the GRU GEMMs
#define EPS 1e-7f

typedef __attribute__((ext_vector_type(2))) float v2f;
typedef __attribute__((ext_vector_type(4))) float v4f;
typedef __attribute__((ext_vector_type(8))) float v8f;

__device__ inline float dot4(v4f a, v4f b) {
  return a.x * b.x + a.y * b.y + a.z * b.z + a.w * b.w;
}

// ---------------------------------------------------------------------------
// utility: block-wide sum (wave32 shuffle + LDS tree)
// ---------------------------------------------------------------------------
__device__ inline float block_reduce_sum(float v, float* red) {
  int tid = threadIdx.x;
  #pragma unroll
  for (int off = 16; off; off >>= 1) v += __shfl_xor(v, off, 32);
  int wid = tid >> 5, lane = tid & 31;
  if (lane == 0) red[wid] = v;
  __syncthreads();
  int nw = blockDim.x >> 5;
  float s = (tid < nw) ? red[tid] : 0.f;
  if (wid == 0) {
    #pragma unroll
    for (int off = 16; off; off >>= 1) s += __shfl_xor(s, off, 32);
    if (lane == 0) red[0] = s;
  }
  __syncthreads();
  float r = red[0];
  __syncthreads();
  return r;
}

// ---------------------------------------------------------------------------
// Kernel S: hop-0 score partials.
// gpart[db][b][t] = sum_{d in chunk} mem[b,t,d] * w_m[0][d]
// grid = (B, NDB), block = 256 (8 waves; wave-per-t, lanes = float4 across d)
// ---------------------------------------------------------------------------
__global__ __launch_bounds__(256)
void score_partial_kernel(const float* __restrict__ mem,
                          const float* __restrict__ wm,   // al_w + 0 (D floats)
                          float* __restrict__ gpart)
{
  const int b   = blockIdx.x;
  const int db  = blockIdx.y;
  const int d0  = db * DCHUNK;
  const int wid = threadIdx.x >> 5, lane = threadIdx.x & 31;

  // per-lane weight slots (slot1 pre-scaled by validity mask)
  const v4f* wm4 = (const v4f*)(wm + d0);
  const int  q1  = (32 + lane < NQ) ? (32 + lane) : (NQ - 1);
  const float s1 = (32 + lane < NQ) ? 1.f : 0.f;
  v4f w0 = wm4[lane];
  v4f w1 = wm4[q1];
  w1.x *= s1; w1.y *= s1; w1.z *= s1; w1.w *= s1;

  for (int t = wid; t < T; t += 8) {
    const v4f* row4 = (const v4f*)(mem + ((size_t)b * T + t) * D + d0);
    if (t + 8 < T) __builtin_prefetch((const float*)row4 + 8 * (size_t)D + 4 * lane, 0, 1);
    v4f m0 = row4[lane];
    v4f m1 = row4[q1];
    float g = dot4(m0, w0) + dot4(m1, w1);
    #pragma unroll
    for (int off = 16; off; off >>= 1) g += __shfl_xor(g, off, 32);
    if (lane == 0) gpart[((size_t)db * B + b) * T + t] = g;
  }
}

// ---------------------------------------------------------------------------
// Kernel A: scalar bias + softmax for hop h (gpart always precomputed).
// grid = B, block = 512 (== T)
// ---------------------------------------------------------------------------
__global__ __launch_bounds__(512)
void attn_scores_kernel(const float* __restrict__ aspect,
                        const unsigned char* __restrict__ mask,
                        const float* __restrict__ al_w,
                        const float* __restrict__ al_b,
                        const float* __restrict__ e,
                        const float* __restrict__ gpart,
                        float* __restrict__ a_out, int hop)
{
  __shared__ float red[32];
  __shared__ float s_scalar;
  const int b   = blockIdx.x;
  const int tid = threadIdx.x;
  const float* wa = al_w + (size_t)hop * AW_STRIDE + D;  // w_a[h] (Da)
  const float* we = wa + DA;                              // w_e[h] (U)

  // scalar part: aspect.w_a + e.w_e + bias
  float p = 0.f;
  for (int j = tid; j < DA; j += blockDim.x) p += aspect[b * DA + j] * wa[j];
  for (int k = tid; k < U;  k += blockDim.x) p += e[b * U + k] * we[k];
  float ssum = block_reduce_sum(p, red);
  if (tid == 0) s_scalar = ssum + al_b[hop];
  __syncthreads();

  const int t = tid;  // blockDim == T
  float g = gpart[b * T + t] + gpart[B * T + b * T + t] +
            gpart[2 * B * T + b * T + t] + s_scalar;
  float val = __expf(g) * (mask[b * T + t] ? 1.f : 0.f);
  float S = block_reduce_sum(val, red);
  a_out[b * T + t] = val / (S + EPS);
}

// ---------------------------------------------------------------------------
// Kernel B: i_al[b,d] = sum_t a[b,t]*mem[b,t,d], fused with next hop's
// score partials gpart[db][b][t] = sum_{d in chunk} mem[b,t,d]*w_m[h+1][d].
// grid = (B, NDB), block = 256 (8 waves; wave-per-t, lanes = float4 across d)
// ---------------------------------------------------------------------------
__global__ __launch_bounds__(256)
void attn_apply_kernel(const float* __restrict__ mem,
                       const float* __restrict__ a_in,
                       const float* __restrict__ wm_next,  // null on last hop
                       float* __restrict__ i_al,
                       float* __restrict__ gpart_next)
{
  __shared__ float sa[T];
  __shared__ v4f   sacc[8 * 64];   // [wave][float4 slot 0..63]
  const int b   = blockIdx.x;
  const int db  = blockIdx.y;
  const int d0  = db * DCHUNK;
  const int tid = threadIdx.x, wid = tid >> 5, lane = tid & 31;

  for (int t = tid; t < T; t += 256) sa[t] = a_in[b * T + t];
  __syncthreads();

  const int  q1  = (32 + lane < NQ) ? (32 + lane) : (NQ - 1);
  const float s1 = (32 + lane < NQ) ? 1.f : 0.f;

  // next-hop weight slots (slot1 pre-scaled so clamped loads contribute 0)
  v4f w0 = {0.f, 0.f, 0.f, 0.f}, w1 = {0.f, 0.f, 0.f, 0.f};
  const bool do_g = (wm_next != nullptr);
  if (do_g) {
    const v4f* wm4 = (const v4f*)(wm_next + d0);
    w0 = wm4[lane];
    w1 = wm4[q1];
    w1.x *= s1; w1.y *= s1; w1.z *= s1; w1.w *= s1;
  }

  v4f acc0 = {0.f, 0.f, 0.f, 0.f}, acc1 = {0.f, 0.f, 0.f, 0.f};

  for (int t = wid; t < T; t += 8) {
    const v4f* row4 = (const v4f*)(mem + ((size_t)b * T + t) * D + d0);
    if (t + 8 < T) __builtin_prefetch((const float*)row4 + 8 * (size_t)D + 4 * lane, 0, 1);
    const float at  = sa[t];
    const float at1 = at * s1;   // mask slot-1 contribution of clamped loads
    v4f m0 = row4[lane];
    v4f m1 = row4[q1];
    acc0.x += at * m0.x;  acc0.y += at * m0.y;
    acc0.z += at * m0.z;  acc0.w += at * m0.w;
    acc1.x += at1 * m1.x; acc1.y += at1 * m1.y;
    acc1.z += at1 * m1.z; acc1.w += at1 * m1.w;
    if (do_g) {
      float g = dot4(m0, w0) + dot4(m1, w1);
      #pragma unroll
      for (int off = 16; off; off >>= 1) g += __shfl_xor(g, off, 32);
      if (lane == 0) gpart_next[((size_t)db * B + b) * T + t] = g;
    }
  }

  // cross-wave reduction of the i_al accumulators (float4 lanes)
  sacc[wid * 64 + lane]      = acc0;
  sacc[wid * 64 + 32 + lane] = acc1;
  __syncthreads();
  if (tid < 64) {
    v4f s = {0.f, 0.f, 0.f, 0.f};
    #pragma unroll
    for (int w = 0; w < 8; w++) {
      v4f v = sacc[w * 64 + tid];
      s.x += v.x; s.y += v.y; s.z += v.z; s.w += v.w;
    }
    if (tid < NQ) *(v4f*)(i_al + (size_t)b * D + d0 + tid * 4) = s;
  }
}

// ---------------------------------------------------------------------------
// WMMA f32 16x16x4 tile GEMM helpers (accumulate into c).
// A: MxK row-major (lda), Bm: KxN row-major (ldb). ISA 7.12.2 fragment layout:
//   A frag: lanes 0-15 hold M rows; lane-half picks K pair {0,1}/{2,3}
//   B frag: lanes 0-15 hold N cols; lane-half picks K pair
// N bounds handled by index-clamp + 0/1 scale: loads stay unconditional and
// EXEC stays all-ones (required for WMMA); no divergent branches in the loop.
// ---------------------------------------------------------------------------
__device__ inline v8f wmma_gemm_acc(const float* __restrict__ A, int lda,
                                    const float* __restrict__ Bm, int ldb,
                                    int m0, int n0, int K, int nmax, v8f c)
{
  const int lane = threadIdx.x & 31;
  const int mr = m0 + (lane & 15);
  const int nc = n0 + (lane & 15);
  const int kh = (lane >> 4) << 1;
  const int ncc   = (nc < nmax) ? nc : (nmax - 1);
  const float nsc = (nc < nmax) ? 1.f : 0.f;
  for (int k = 0; k < K; k += 4) {
    v2f a = *(const v2f*)(A + (size_t)mr * lda + k + kh);   // 8B-aligned (lda,k,kh even)
    v2f bf;
    bf.x = Bm[(size_t)(k + kh) * ldb + ncc] * nsc;
    bf.y = Bm[(size_t)(k + kh + 1) * ldb + ncc] * nsc;
    c = __builtin_amdgcn_wmma_f32_16x16x4_f32(false, a, false, bf,
                                              (short)0, c, false, false);
  }
  return c;
}

// Same, but A is an elementwise product A1⊙A2 (for (r*e)@Wg)
__device__ inline v8f wmma_gemm_acc_had(const float* __restrict__ A1,
                                        const float* __restrict__ A2, int lda,
                                        const float* __restrict__ Bm, int ldb,
                                        int m0, int n0, int K, int nmax, v8f c)
{
  const int lane = threadIdx.x & 31;
  const int mr = m0 + (lane & 15);
  const int nc = n0 + (lane & 15);
  const int kh = (lane >> 4) << 1;
  const int ncc   = (nc < nmax) ? nc : (nmax - 1);
  const float nsc = (nc < nmax) ? 1.f : 0.f;
  for (int k = 0; k < K; k += 4) {
    v2f a1 = *(const v2f*)(A1 + (size_t)mr * lda + k + kh);
    v2f a2 = *(const v2f*)(A2 + (size_t)mr * lda + k + kh);
    v2f a; a.x = a1.x * a2.x; a.y = a1.y * a2.y;
    v2f bf;
    bf.x = Bm[(size_t)(k + kh) * ldb + ncc] * nsc;
    bf.y = Bm[(size_t)(k + kh + 1) * ldb + ncc] * nsc;
    c = __builtin_amdgcn_wmma_f32_16x16x4_f32(false, a, false, bf,
                                              (short)0, c, false, false);
  }
  return c;
}

// ---------------------------------------------------------------------------
// Kernel C1: r = sigmoid(i_al@Wr + e@Ur), z = sigmoid(i_al@Wz + e@Uz)
// grid = (NT, 2), block = 128 (4 waves; wave = m-tile of 16 rows)
// ---------------------------------------------------------------------------
__global__ __launch_bounds__(128)
void gru_rz_kernel(const float* __restrict__ i_al, const float* __restrict__ e,
                   const float* __restrict__ Wr, const float* __restrict__ Ur,
                   const float* __restrict__ Wz, const float* __restrict__ Uz,
                   float* __restrict__ r_out, float* __restrict__ z_out)
{
  const float* W  = blockIdx.y ? Wz : Wr;
  const float* Um = blockIdx.y ? Uz : Ur;
  float* out      = blockIdx.y ? z_out : r_out;

  const int lane = threadIdx.x & 31;
  const int m0 = (threadIdx.x >> 5) * 16;
  const int n0 = blockIdx.x * 16;

  v8f c = {};
  c = wmma_gemm_acc(i_al, D, W, U, m0, n0, D, U, c);
  c = wmma_gemm_acc(e,    U, Um, U, m0, n0, U, U, c);

  const int col   = n0 + (lane & 15);
  const int rbase = m0 + ((lane >> 4) ? 8 : 0);
  if (col < U) {
    #pragma unroll
    for (int v = 0; v < 8; v++) {
      float x = c[v];
      out[(size_t)(rbase + v) * U + col] = 1.f / (1.f + __expf(-x));
    }
  }
}

// ---------------------------------------------------------------------------
// Kernel C2: e_tilde = tanh(i_al@Wx + (r⊙e)@Wg); e' = (1-z)*e + z*e_tilde
// grid = NT, block = 128
// ---------------------------------------------------------------------------
__global__ __launch_bounds__(128)
void gru_update_kernel(const float* __restrict__ i_al, const float* __restrict__ e,
                       const float* __restrict__ r, const float* __restrict__ z,
                       const float* __restrict__ Wx, const float* __restrict__ Wg,
                       float* __restrict__ e_next)
{
  const int lane = threadIdx.x & 31;
  const int m0 = (threadIdx.x >> 5) * 16;
  const int n0 = blockIdx.x * 16;

  v8f c = {};
  c = wmma_gemm_acc(i_al, D, Wx, U, m0, n0, D, U, c);
  c = wmma_gemm_acc_had(r, e, U, Wg, U, m0, n0, U, U, c);

  const int col   = n0 + (lane & 15);
  const int rbase = m0 + ((lane >> 4) ? 8 : 0);
  if (col < U) {
    #pragma unroll
    for (int v = 0; v < 8; v++) {
      size_t idx = (size_t)(rbase + v) * U + col;
      float et = tanhf(c[v]);
      float zz = z[idx];
      e_next[idx] = (1.f - zz) * e[idx] + zz * et;
    }
  }
}

// ---------------------------------------------------------------------------
// tiny utility kernels
// ---------------------------------------------------------------------------
__global__ void zero_kernel(float* p, int n) {
  int i = blockIdx.x * blockDim.x + threadIdx.x;
  if (i < n) p[i] = 0.f;
}
__global__ void copy_kernel(const float* __restrict__ s, float* __restrict__ d, int n) {
  int i = blockIdx.x * blockDim.x + threadIdx.x;
  if (i < n) d[i] = s[i];
}

// ---------------------------------------------------------------------------
// host launcher
// ---------------------------------------------------------------------------
extern "C" void kernel_launch(void* const* d_in, const int* in_sizes, int n_in,
                              void* d_out, int out_size, void* d_ws, size_t ws_size,
                              hipStream_t stream) {
  const float* mem    = (const float*)d_in[0];
  const float* aspect = (const float*)d_in[1];
  const unsigned char* mask = (const unsigned char*)d_in[2];
  const float* al_w   = (const float*)d_in[3];
  const float* al_b   = (const float*)d_in[4];
  const float* wr     = (const float*)d_in[5];
  const float* ur     = (const float*)d_in[6];
  const float* wz     = (const float*)d_in[7];
  const float* uz     = (const float*)d_in[8];
  const float* wx     = (const float*)d_in[9];
  const float* wg     = (const float*)d_in[10];

  float* ws = (float*)d_ws;
  float* ebuf0 = ws;                       // B*U
  float* ebuf1 = ebuf0 + B * U;            // B*U
  float* a_buf = ebuf1 + B * U;            // B*T
  float* ial   = a_buf + B * T;            // B*D
  float* rbuf  = ial + B * D;              // B*U
  float* zbuf  = rbuf + B * U;             // B*U
  float* gpart = zbuf + B * U;             // NDB*B*T
  float* ebuf[2] = { ebuf0, ebuf1 };

  zero_kernel<<<(B * U + 255) / 256, 256, 0, stream>>>(ebuf[0], B * U);

  // hop-0 score partials (full streaming pass over memory)
  score_partial_kernel<<<dim3(B, NDB), 256, 0, stream>>>(mem, al_w, gpart);

  for (int h = 0; h < H; h++) {
    const float* e_cur = ebuf[h & 1];
    float*       e_nxt = ebuf[(h + 1) & 1];

    attn_scores_kernel<<<B, 512, 0, stream>>>(
        aspect, mask, al_w, al_b, e_cur, gpart, a_buf, h);

    const float* wmn = (h + 1 < H) ? (al_w + (size_t)(h + 1) * AW_STRIDE) : nullptr;
    attn_apply_kernel<<<dim3(B, NDB), 256, 0, stream>>>(
        mem, a_buf, wmn, ial, gpart);

    gru_rz_kernel<<<dim3(NT, 2), 128, 0, stream>>>(
        ial, e_cur, wr, ur, wz, uz, rbuf, zbuf);

    gru_update_kernel<<<NT, 128, 0, stream>>>(
        ial, e_cur, rbuf, zbuf, wx, wg, e_nxt);
  }

  copy_kernel<<<(B * U + 255) / 256, 256, 0, stream>>>(ebuf[H & 1], (float*)d_out, B * U);
}